// AsyncAlignmentModule_4114578670415
// MI455X (gfx1250) — compile-verified
//
#include <hip/hip_runtime.h>
#include <stdint.h>

// ---------------------------------------------------------------------------
// AsyncAlignmentModule for MI455X (gfx1250).
//
// Shapes (fixed by reference): B=16, C=128, Ta=R=2048, Tb=1024.
// Output layout (float32, concatenated flat in return order):
//   aligned       [2,B,C,R]  : 2*B*C*R elements
//   aligned_masks [2,B,R]    : 2*B*R
//   src_idx       [2,B,R]    : 2*B*R   (-1..S-1, exact in fp32)
//   valid_ratio   [2,B]      : 2*B
//
// Bandwidth-bound (~60 MB traffic -> ~2.6us floor @ 23.3 TB/s). CDNA5 path:
// async global->LDS staging (global_load_async_to_lds_b128 / ASYNCcnt /
// s_wait_asynccnt) so the B*R*S argmin inner loop reads LDS, and the
// reference axis is tiled across 256 workgroups to fill the chip.
// ---------------------------------------------------------------------------

#define BIG_DIST 1e30f
#define TILE_R 256  // refs per block in the argmin pass

// Async DMA of 16 bytes from global (saddr + 32-bit voffset, GVS mode) into
// LDS at lds_addr. Tracked by ASYNCcnt.
__device__ __forceinline__ void async_copy_b128(uint32_t lds_addr,
                                                const void* sbase,
                                                uint32_t voff) {
  asm volatile("global_load_async_to_lds_b128 %0, %1, %2"
               :
               : "v"(lds_addr), "v"(voff), "s"(sbase)
               : "memory");
}

__device__ __forceinline__ void wait_asynccnt0() {
  asm volatile("s_wait_asynccnt 0x0" ::: "memory");
}

// Grid: (R/TILE_R, B). Each block stages the full src row (timestamps+masks,
// <=16KB) into LDS via async DMA, then each thread computes the masked
// nearest-neighbor argmin for one reference timestamp (first-min tie-break,
// matching jnp.argmin).
__global__ void __launch_bounds__(TILE_R)
align_argmin_kernel(const float* __restrict__ ts_src,
                    const float* __restrict__ mask_src,
                    const float* __restrict__ ts_ref,
                    const float* __restrict__ mask_ref,
                    float* __restrict__ out_mask,   // [B,R]
                    float* __restrict__ out_idx,    // [B,R] (float-encoded)
                    int R, int S) {
  __shared__ __align__(16) float lds_t[2048];
  __shared__ __align__(16) float lds_m[2048];

  const int b = blockIdx.y;
  const int tid = threadIdx.x;

  // ---- async stage this batch row's src timestamps + masks into LDS ----
  const float* tsrc = ts_src + (size_t)b * S;
  const float* msrc = mask_src + (size_t)b * S;
  const int nvec = S >> 2;  // 16-byte chunks
  for (int i = tid; i < nvec; i += TILE_R) {
    uint32_t la_t = (uint32_t)(size_t)(&lds_t[i << 2]);
    uint32_t la_m = (uint32_t)(size_t)(&lds_m[i << 2]);
    uint32_t off = (uint32_t)(i << 4);
    async_copy_b128(la_t, (const void*)tsrc, off);
    async_copy_b128(la_m, (const void*)msrc, off);
  }
  wait_asynccnt0();   // drain this wave's ASYNCcnt
  __syncthreads();    // all waves' LDS writes visible

  // ---- masked nearest-neighbor argmin over LDS, float4-vectorized ----
  const int r = blockIdx.x * TILE_R + tid;  // R is a multiple of TILE_R
  const size_t ref_off = (size_t)b * R + r;
  const float rt = ts_ref[ref_off];
  const float rm = mask_ref[ref_off];

  const float4* t4 = (const float4*)lds_t;
  const float4* m4 = (const float4*)lds_m;
  const int S4 = S >> 2;

  float best = BIG_DIST;
  int bi = 0;
  for (int s4 = 0; s4 < S4; ++s4) {
    const float4 tv = t4[s4];
    const float4 mv = m4[s4];
    const float d0 = (mv.x > 0.f) ? fabsf(rt - tv.x) : BIG_DIST;
    const float d1 = (mv.y > 0.f) ? fabsf(rt - tv.y) : BIG_DIST;
    const float d2 = (mv.z > 0.f) ? fabsf(rt - tv.z) : BIG_DIST;
    const float d3 = (mv.w > 0.f) ? fabsf(rt - tv.w) : BIG_DIST;
    const int base = s4 << 2;
    if (d0 < best) { best = d0; bi = base; }
    if (d1 < best) { best = d1; bi = base + 1; }
    if (d2 < best) { best = d2; bi = base + 2; }
    if (d3 < best) { best = d3; bi = base + 3; }
  }
  // ok = ref valid && at least one valid source (any valid src gives
  // best <= 200 << BIG; best stays BIG only if all sources are masked out)
  const bool ok = (rm > 0.f) && (best < BIG_DIST);
  out_mask[ref_off] = ok ? 1.f : 0.f;
  out_idx[ref_off] = ok ? (float)bi : -1.f;
}

// One block per (modality,batch) row: mean of the already-written mask row.
// Integer counts summed in LDS -> exact and order-independent.
__global__ void __launch_bounds__(256)
ratio_kernel(const float* __restrict__ mask,  // [2*B, R] contiguous
             float* __restrict__ ratio,       // [2*B]
             int R) {
  __shared__ int s_cnt;
  if (threadIdx.x == 0) s_cnt = 0;
  __syncthreads();

  const float4* row = (const float4*)(mask + (size_t)blockIdx.x * R);
  int c = 0;
  for (int i = threadIdx.x; i < (R >> 2); i += blockDim.x) {
    const float4 v = row[i];
    c += (v.x > 0.f) + (v.y > 0.f) + (v.z > 0.f) + (v.w > 0.f);
  }
  atomicAdd(&s_cnt, c);
  __syncthreads();
  if (threadIdx.x == 0) ratio[blockIdx.x] = (float)s_cnt / (float)R;
}

// Streaming gather: each thread emits 4 consecutive r outputs for one (b,c).
__global__ void __launch_bounds__(256)
gather_kernel(const float* __restrict__ vals,  // [B,C,S]
              const float* __restrict__ idx,   // [B,R] float-encoded indices
              float* __restrict__ out,         // [B,C,R]
              int B, int C, int R, int S) {
  const int t = blockIdx.x * blockDim.x + threadIdx.x;
  const int R4 = R >> 2;
  const int total = B * C * R4;
  if (t >= total) return;

  const int r4 = t % R4;
  const int c = (t / R4) % C;
  const int b = t / (R4 * C);

  const float4 idf = ((const float4*)(idx + (size_t)b * R))[r4];
  const int i0 = (int)idf.x;
  const int i1 = (int)idf.y;
  const int i2 = (int)idf.z;
  const int i3 = (int)idf.w;

  const float* vrow = vals + ((size_t)b * C + c) * S;
  float4 o;
  o.x = (i0 >= 0) ? vrow[i0] : 0.f;
  o.y = (i1 >= 0) ? vrow[i1] : 0.f;
  o.z = (i2 >= 0) ? vrow[i2] : 0.f;
  o.w = (i3 >= 0) ? vrow[i3] : 0.f;
  ((float4*)(out + ((size_t)b * C + c) * R))[r4] = o;
}

extern "C" void kernel_launch(void* const* d_in, const int* in_sizes, int n_in,
                              void* d_out, int out_size, void* d_ws, size_t ws_size,
                              hipStream_t stream) {
  (void)in_sizes; (void)n_in; (void)out_size; (void)d_ws; (void)ws_size;

  const float* values_a = (const float*)d_in[0];  // [B,C,TA]
  const float* ts_a     = (const float*)d_in[1];  // [B,TA]
  const float* mask_a   = (const float*)d_in[2];  // [B,TA]
  const float* values_b = (const float*)d_in[3];  // [B,C,TB]
  const float* ts_b     = (const float*)d_in[4];  // [B,TB]
  const float* mask_b   = (const float*)d_in[5];  // [B,TB]

  const int B = 16, C = 128, TA = 2048, TB = 1024;
  const int R = TA;  // modality A defines the reference timeline

  float* out = (float*)d_out;
  const size_t al_sz = (size_t)B * C * R;  // per-modality aligned block
  const size_t mk_sz = (size_t)B * R;      // per-modality mask/idx block

  float* aligned0 = out;
  float* aligned1 = out + al_sz;
  float* omask    = out + 2 * al_sz;       // [2*B, R] contiguous
  float* omask0   = omask;
  float* omask1   = omask + mk_sz;
  float* oidx0    = out + 2 * al_sz + 2 * mk_sz;
  float* oidx1    = oidx0 + mk_sz;
  float* oratio   = out + 2 * al_sz + 4 * mk_sz;  // [2*B] contiguous

  // Pass 1: masked nearest-neighbor argmin, async-LDS staged, R tiled across
  // blocks for full-chip occupancy (2 * 8 * 16 = 256 workgroups).
  dim3 ag(R / TILE_R, B);
  align_argmin_kernel<<<ag, TILE_R, 0, stream>>>(ts_a, mask_a, ts_a, mask_a,
                                                 omask0, oidx0, R, TA);
  align_argmin_kernel<<<ag, TILE_R, 0, stream>>>(ts_b, mask_b, ts_a, mask_a,
                                                 omask1, oidx1, R, TB);

  // Pass 2: valid_ratio = row-mean of masks (deterministic integer reduce).
  ratio_kernel<<<2 * B, 256, 0, stream>>>(omask, oratio, R);

  // Pass 3: channel gather (bandwidth-bound streaming).
  const int total4 = B * C * (R >> 2);
  const int blocks = (total4 + 255) / 256;
  gather_kernel<<<blocks, 256, 0, stream>>>(values_a, oidx0, aligned0, B, C, R, TA);
  gather_kernel<<<blocks, 256, 0, stream>>>(values_b, oidx1, aligned1, B, C, R, TB);
}